// ObsOpt_89489938579932
// MI455X (gfx1250) — compile-verified
//
#include <hip/hip_runtime.h>
#include <math.h>

#define N_PROB       4096
#define M_OBS        1024
#define M_TOT        1030
#define T_PER_LANE   33      // ceil(1030/32) constraints per lane
#define IP_ITERS     30
#define SIGMA_C      0.1f
#define P_DELTA_C    5.0f
#define INPUT_LIM    0.05f
#define WAVES_PER_BLOCK 8
#define BLOCK        256

typedef float v2f __attribute__((ext_vector_type(2)));
typedef float v8f __attribute__((ext_vector_type(8)));

// ---------------------------------------------------------------------------
// Full-precision 32-lane sum on the matrix pipe (V_WMMA_F32_16X16X4_F32).
// Stage 1: D1 = V(16x4) * ones(4x16)  -> D1[m][n] = v[m] + v[m+16]  (all n)
//          (A slots: VGPR0 carries all 32 lane values, VGPR1 = 0; any
//           bijective slot mapping yields row sums covering each lane once.)
// Stage 2: D2 += ones(16x4) * B(4x16) chained over the 4 VGPR-pairs of D1.
//          Each D1 VGPR holds {rowsum(p), rowsum(p+8)} in its lane halves
//          (documented C/D layout); B rows are striped per half-VGPR, so each
//          column sums 4 distinct row-sums -> after 4 chained WMMAs every
//          element of D2 equals the full 32-lane sum (order-independent).
// ---------------------------------------------------------------------------
static __device__ __forceinline__ float wave_sum32(float v) {
  v2f a;    a[0] = v;    a[1] = 0.0f;
  v2f ones; ones[0] = 1.0f; ones[1] = 1.0f;
  v8f zero8 = {0.f,0.f,0.f,0.f,0.f,0.f,0.f,0.f};
  v8f d1 = __builtin_amdgcn_wmma_f32_16x16x4_f32(false, a, false, ones,
                                                 (short)0, zero8, false, false);
  v8f acc = {0.f,0.f,0.f,0.f,0.f,0.f,0.f,0.f};
  v2f b;
  b[0] = d1[0]; b[1] = d1[1];
  acc = __builtin_amdgcn_wmma_f32_16x16x4_f32(false, ones, false, b,
                                              (short)0, acc, false, false);
  b[0] = d1[2]; b[1] = d1[3];
  acc = __builtin_amdgcn_wmma_f32_16x16x4_f32(false, ones, false, b,
                                              (short)0, acc, false, false);
  b[0] = d1[4]; b[1] = d1[5];
  acc = __builtin_amdgcn_wmma_f32_16x16x4_f32(false, ones, false, b,
                                              (short)0, acc, false, false);
  b[0] = d1[6]; b[1] = d1[7];
  acc = __builtin_amdgcn_wmma_f32_16x16x4_f32(false, ones, false, b,
                                              (short)0, acc, false, false);
  return acc[0];   // wave-uniform
}

static __device__ __forceinline__ float wave_min32(float v) {
  #pragma unroll
  for (int off = 16; off > 0; off >>= 1)
    v = fminf(v, __shfl_xor(v, off, 32));
  return v;        // wave-uniform
}

// ---------------------------------------------------------------------------
// Main kernel: one wave32 per QP problem.
// ---------------------------------------------------------------------------
__global__ __launch_bounds__(BLOCK)
void qp_solve_kernel(const float* __restrict__ xcur,
                     const float* __restrict__ xtgt,
                     const float* __restrict__ xobs,
                     const float* __restrict__ robs,
                     const float* __restrict__ lbd_p,
                     const float* __restrict__ alpha_p,
                     float* __restrict__ out_u,
                     float* __restrict__ ws_loss) {
  __shared__ float2 obs[M_OBS];
  __shared__ float  orad[M_OBS];

  const int tid = threadIdx.x;
  for (int k = tid; k < M_OBS; k += BLOCK) {
    obs[k]  = ((const float2*)xobs)[k];
    orad[k] = robs[k];
  }
  __syncthreads();

  const int lane = tid & 31;
  const int n    = blockIdx.x * WAVES_PER_BLOCK + (tid >> 5);

  const float lbd  = lbd_p[0];
  const float alph = alpha_p[0];
  const float cx = xcur[2*n],   cy = xcur[2*n+1];
  const float tx = xtgt[2*n],   ty = xtgt[2*n+1];

  // --- constraint row generator: A(j,:) and b(j) ---------------------------
  auto eval_con = [&](int j, float& a0, float& a1, float& a2, float& bb) {
    if (j == 0) {                       // CLF row
      float dx = cx - tx, dy = cy - ty;
      a0 = 2.f*dx; a1 = 2.f*dy; a2 = -1.f;
      bb = -lbd * (dx*dx + dy*dy);
    } else if (j <= M_OBS) {            // CBF rows:  A = -dir, b = alpha*(dist-r)
      float2 o = obs[j-1];
      float dx = cx - o.x, dy = cy - o.y;
      float dist = sqrtf(dx*dx + dy*dy);
      float inv  = 1.0f / (dist + 1e-6f);
      a0 = -dx * inv; a1 = -dy * inv; a2 = 0.f;
      bb = alph * (dist - orad[j-1]);
    } else if (j == M_OBS+1) { a0= 1.f; a1= 0.f; a2=0.f;  bb = INPUT_LIM; }
      else if (j == M_OBS+2) { a0=-1.f; a1= 0.f; a2=0.f;  bb = INPUT_LIM; }
      else if (j == M_OBS+3) { a0= 0.f; a1= 1.f; a2=0.f;  bb = INPUT_LIM; }
      else if (j == M_OBS+4) { a0= 0.f; a1=-1.f; a2=0.f;  bb = INPUT_LIM; }
      else if (j == M_OBS+5) { a0= 0.f; a1= 0.f; a2=-1.f; bb = 0.f; }
      else                   { a0= 0.f; a1= 0.f; a2=0.f;  bb = 0.f; }   // inactive
  };

  // --- collision loss partial: sum_j relu(r_j - dist_j) --------------------
  float loss = 0.f;
  #pragma unroll 4
  for (int t = 0; t < 32; ++t) {
    int o = t*32 + lane;
    float dx = cx - obs[o].x, dy = cy - obs[o].y;
    float dist = sqrtf(dx*dx + dy*dy);
    loss += fmaxf(orad[o] - dist, 0.f);
  }
  loss = wave_sum32(loss);
  if (lane == 0) ws_loss[n] = loss;

  // --- interior point state ------------------------------------------------
  float s[T_PER_LANE], lm[T_PER_LANE], dsv[T_PER_LANE];
  #pragma unroll
  for (int t = 0; t < T_PER_LANE; ++t) { s[t] = 1.f; lm[t] = 1.f; }
  float z0 = 0.f, z1 = 0.f, z2 = 0.f;

  #pragma unroll 1
  for (int it = 0; it < IP_ITERS; ++it) {
    // ---- sweep 1: mu = SIGMA * mean(s*lam) -------------------------------
    float sl = 0.f;
    #pragma unroll
    for (int t = 0; t < T_PER_LANE; ++t) {
      int j = t*32 + lane;
      sl += (j < M_TOT) ? s[t]*lm[t] : 0.f;
    }
    float mu = SIGMA_C * wave_sum32(sl) * (1.0f / (float)M_TOT);

    // ---- sweep 2: H = Q + A^T D A,  g = lam + (lam*r_p - r_c)/s ----------
    float hxx=0.f,hxy=0.f,hxz=0.f,hyy=0.f,hyz=0.f,hzz=0.f;
    float gx=0.f, gy=0.f, gz=0.f;
    #pragma unroll
    for (int t = 0; t < T_PER_LANE; ++t) {
      int j = t*32 + lane;
      float a0,a1,a2,bb; eval_con(j, a0,a1,a2,bb);
      float Az   = a0*z0 + a1*z1 + a2*z2;
      float rp   = Az + s[t] - bb;
      float rc   = lm[t]*s[t] - mu;
      float invs = 1.f / s[t];
      float D    = lm[t]*invs;
      float g    = lm[t] + (lm[t]*rp - rc)*invs;
      float m    = (j < M_TOT) ? 1.f : 0.f;
      D *= m; g *= m;
      hxx += D*a0*a0; hxy += D*a0*a1; hxz += D*a0*a2;
      hyy += D*a1*a1; hyz += D*a1*a2; hzz += D*a2*a2;
      gx  += g*a0;    gy  += g*a1;    gz  += g*a2;
    }
    hxx = wave_sum32(hxx); hxy = wave_sum32(hxy); hxz = wave_sum32(hxz);
    hyy = wave_sum32(hyy); hyz = wave_sum32(hyz); hzz = wave_sum32(hzz);
    gx  = wave_sum32(gx);  gy  = wave_sum32(gy);  gz  = wave_sum32(gz);

    hxx += 1.f; hyy += 1.f; hzz += P_DELTA_C;          // H = Q + ...
    float rx = -(z0 + gx);
    float ry = -(z1 + gy);
    float rz = -(P_DELTA_C*z2 + gz);                   // rhs = -(Qz + A^T g)

    // ---- 3x3 SPD solve by adjugate (wave-uniform, every lane) ------------
    float c00 = hyy*hzz - hyz*hyz;
    float c01 = hxz*hyz - hxy*hzz;
    float c02 = hxy*hyz - hxz*hyy;
    float det = hxx*c00 + hxy*c01 + hxz*c02;
    float c11 = hxx*hzz - hxz*hxz;
    float c12 = hxy*hxz - hxx*hyz;
    float c22 = hxx*hyy - hxy*hxy;
    float idet = 1.f / det;
    float dz0 = (c00*rx + c01*ry + c02*rz) * idet;
    float dz1 = (c01*rx + c11*ry + c12*rz) * idet;
    float dz2 = (c02*rx + c12*ry + c22*rz) * idet;

    // ---- sweep 3: ds, dlam, fraction-to-boundary ratios ------------------
    float rmin = 1e9f;
    #pragma unroll
    for (int t = 0; t < T_PER_LANE; ++t) {
      int j = t*32 + lane;
      float a0,a1,a2,bb; eval_con(j, a0,a1,a2,bb);
      float Az  = a0*z0 + a1*z1 + a2*z2;
      float rp  = Az + s[t] - bb;
      float d_s = -rp - (a0*dz0 + a1*dz1 + a2*dz2);
      dsv[t] = d_s;
      float rc  = lm[t]*s[t] - mu;
      float d_l = (-rc - lm[t]*d_s) / s[t];
      bool act  = (j < M_TOT);
      float r1 = (act && d_s < 0.f) ? (-s[t]  / d_s) : 1e9f;
      float r2 = (act && d_l < 0.f) ? (-lm[t] / d_l) : 1e9f;
      rmin = fminf(rmin, fminf(r1, r2));
    }
    rmin = wave_min32(rmin);
    float astep = fminf(1.f, 0.99f * rmin);

    // ---- sweep 4: apply step (inactive slots frozen at s=lam=1) ----------
    #pragma unroll
    for (int t = 0; t < T_PER_LANE; ++t) {
      int j = t*32 + lane;
      if (j < M_TOT) {
        float rc  = lm[t]*s[t] - mu;
        float d_l = (-rc - lm[t]*dsv[t]) / s[t];
        s[t]  += astep * dsv[t];
        lm[t] += astep * d_l;
      }
    }
    z0 += astep*dz0; z1 += astep*dz1; z2 += astep*dz2;
  }

  if (lane == 0) {
    out_u[2*n]   = z0;   // u_pred = z[:, :2]
    out_u[2*n+1] = z1;
  }
}

// ---------------------------------------------------------------------------
// Deterministic fixed-order reduction of the 4096 per-problem loss partials.
// ---------------------------------------------------------------------------
__global__ void loss_reduce_kernel(const float* __restrict__ ws_loss,
                                   float* __restrict__ out_loss) {
  __shared__ float sh[128];
  float v = 0.f;
  for (int i = threadIdx.x; i < N_PROB; i += 128) v += ws_loss[i];
  sh[threadIdx.x] = v;
  __syncthreads();
  if (threadIdx.x == 0) {
    float tot = 0.f;
    #pragma unroll 1
    for (int i = 0; i < 128; ++i) tot += sh[i];
    out_loss[0] = tot;
  }
}

extern "C" void kernel_launch(void* const* d_in, const int* in_sizes, int n_in,
                              void* d_out, int out_size, void* d_ws, size_t ws_size,
                              hipStream_t stream) {
  (void)in_sizes; (void)n_in; (void)out_size; (void)ws_size;
  const float* xcur  = (const float*)d_in[0];  // (4096,2)
  const float* xtgt  = (const float*)d_in[1];  // (4096,2)
  const float* xobs  = (const float*)d_in[2];  // (1024,2)
  const float* robs  = (const float*)d_in[3];  // (1024,)
  const float* lbd   = (const float*)d_in[4];  // (1,)
  const float* alph  = (const float*)d_in[5];  // (1,)
  float* out   = (float*)d_out;                // [0..8191]=u_pred, [8192]=loss
  float* wsl   = (float*)d_ws;                 // 4096 per-wave loss partials

  qp_solve_kernel<<<N_PROB / WAVES_PER_BLOCK, BLOCK, 0, stream>>>(
      xcur, xtgt, xobs, robs, lbd, alph, out, wsl);
  loss_reduce_kernel<<<1, 128, 0, stream>>>(wsl, out + 2 * N_PROB);
}